// TemporalSmoothing_5291399708732
// MI455X (gfx1250) — compile-verified
//
#include <hip/hip_runtime.h>

typedef __attribute__((ext_vector_type(2))) float v2f;
typedef __attribute__((ext_vector_type(8))) float v8f;

#define BATCH 16
#define SEQ   8192
#define FEAT  128
#define NSEG  64
#define SEGLEN (SEQ / NSEG)     // 128
#define CHUNKS (SEGLEN / 16)    // 8
#define NFT    (FEAT / 16)      // 8 feature tiles
#define WAVES  (BATCH * NFT * NSEG)  // 8192 waves

// uniform-trip-count integer power (n in [0,16])
__device__ __forceinline__ float powi_f(float w, int n) {
    float r = 1.0f;
    #pragma unroll
    for (int i = 0; i < 16; ++i) r = (i < n) ? r * w : r;
    return r;
}

// ---------------------------------------------------------------------------
// Kernel 1: per-segment EWMA scan via chained V_WMMA_F32_16X16X4_F32.
// Each wave handles one (batch, 16-feature tile, 128-step segment).
// D(16x16) = L(16x16, lower-tri alpha*w^(m-k)) * X(16 time x 16 feat) + C,
// where C[j] = w^(M+1) * carry (per-lane feature carry).
//
// A layout (ISA 16x4 f32): lane M = lane&15; V0 holds K = 2*(lane>>4),
//                          V1 holds K = 2*(lane>>4)+1 (per 4-wide K slice).
// B layout (4x16 f32, assumed symmetric): lane N = lane&15;
//                          V0 row K = 2*(lane>>4), V1 row K = 2*(lane>>4)+1.
// C/D layout (ISA): lane N = lane&15, VGPR j holds M = j + 8*(lane>>4).
// ---------------------------------------------------------------------------
__global__ void __launch_bounds__(128)
ewma_scan_kernel(const float* __restrict__ x, const float* __restrict__ alpha_p,
                 float* __restrict__ out, float* __restrict__ carries)
{
    const int lane = threadIdx.x & 31;
    const int widx = blockIdx.x * (blockDim.x >> 5) + (threadIdx.x >> 5);
    const int seg = widx % NSEG;
    const int ft  = (widx / NSEG) % NFT;
    const int b   = widx / (NSEG * NFT);

    const float alpha = alpha_p[0];
    const float w = 1.0f - alpha;

    const int half = lane >> 4;   // which half-wave
    const int m    = lane & 15;   // M for A, N (feature col) for B/C/D
    const int fb   = ft * 16;

    // A-matrix K-slices (constant per wave): L[m,k] = (k<=m) ? alpha*w^(m-k) : 0
    v2f A[4];
    #pragma unroll
    for (int q = 0; q < 4; ++q) {
        const int c0 = 4 * q + 2 * half;
        const int c1 = c0 + 1;
        A[q].x = (c0 <= m) ? alpha * powi_f(w, m - c0) : 0.0f;
        A[q].y = (c1 <= m) ? alpha * powi_f(w, m - c1) : 0.0f;
    }
    // carry decay per accumulator row: dec[j] = w^(j + 8*half + 1)
    float dec[8];
    {
        float p = powi_f(w, 8 * half + 1);
        #pragma unroll
        for (int j = 0; j < 8; ++j) { dec[j] = p; p *= w; }
    }

    const size_t xbase_bf = (size_t)b * SEQ * FEAT + fb + m;
    // s_0 = x_0 is equivalent to virtual carry s_{-1} = x_0 (alpha*x0 + w*x0 = x0)
    float carry = (seg == 0) ? x[xbase_bf] : 0.0f;

    for (int c = 0; c < CHUNKS; ++c) {
        const int t0 = seg * SEGLEN + c * 16;
        const size_t base = xbase_bf + (size_t)t0 * FEAT;

        v2f Bq[4];
        #pragma unroll
        for (int q = 0; q < 4; ++q) {
            const int r0 = 4 * q + 2 * half;
            Bq[q].x = x[base + (size_t)r0 * FEAT];
            Bq[q].y = x[base + (size_t)(r0 + 1) * FEAT];
        }

        v8f acc;
        #pragma unroll
        for (int j = 0; j < 8; ++j) acc[j] = dec[j] * carry;

        acc = __builtin_amdgcn_wmma_f32_16x16x4_f32(false, A[0], false, Bq[0], (short)0, acc, false, false);
        acc = __builtin_amdgcn_wmma_f32_16x16x4_f32(false, A[1], false, Bq[1], (short)0, acc, false, false);
        acc = __builtin_amdgcn_wmma_f32_16x16x4_f32(false, A[2], false, Bq[2], (short)0, acc, false, false);
        acc = __builtin_amdgcn_wmma_f32_16x16x4_f32(false, A[3], false, Bq[3], (short)0, acc, false, false);

        #pragma unroll
        for (int j = 0; j < 8; ++j)
            out[base + (size_t)(j + 8 * half) * FEAT] = acc[j];

        // new carry = D row M=15 (VGPR 7 of lanes 16..31), broadcast per feature
        carry = __shfl(acc[7], m + 16, 32);
    }

    if (half == 0)
        carries[(((size_t)b * NFT + ft) * NSEG + seg) * 16 + m] = carry;
}

// ---------------------------------------------------------------------------
// Kernel 2: sequential combine of 64 segment carries per (b, f).
// S_0 = carries[0] (seed already included); for s>=1:
//   seg_in[s] = S_{s-1};  S_s = w^128 * S_{s-1} + carries[s]
// ---------------------------------------------------------------------------
__global__ void __launch_bounds__(256)
ewma_combine_kernel(const float* __restrict__ carries, const float* __restrict__ alpha_p,
                    float* __restrict__ seg_in)
{
    const int tid = blockIdx.x * blockDim.x + threadIdx.x;
    if (tid >= BATCH * FEAT) return;
    const int f = tid % FEAT;
    const int b = tid / FEAT;
    const int ft = f >> 4, fi = f & 15;

    const float w = 1.0f - alpha_p[0];
    float wL = w;                      // w^128 = w^(2^7)
    #pragma unroll
    for (int i = 0; i < 7; ++i) wL *= wL;

    float S = 0.0f;
    for (int s = 0; s < NSEG; ++s) {
        const size_t idx = (((size_t)b * NFT + ft) * NSEG + s) * 16 + fi;
        seg_in[idx] = (s == 0) ? 0.0f : S;
        const float c = carries[idx];
        S = (s == 0) ? c : wL * S + c;
    }
}

// ---------------------------------------------------------------------------
// Kernel 3: fix-up. out[t0+M] += w^(M+1) * (w^(16c) * s_in). Exact early-exit:
// once the carry term underflows to exactly 0 in fp32, remaining adds are
// no-ops, so the wave breaks.
// ---------------------------------------------------------------------------
__global__ void __launch_bounds__(128)
ewma_fixup_kernel(const float* __restrict__ seg_in, const float* __restrict__ alpha_p,
                  float* __restrict__ out)
{
    const int lane = threadIdx.x & 31;
    const int widx = blockIdx.x * (blockDim.x >> 5) + (threadIdx.x >> 5);
    const int seg = widx % NSEG;
    const int ft  = (widx / NSEG) % NFT;
    const int b   = widx / (NSEG * NFT);
    const int half = lane >> 4;
    const int m    = lane & 15;
    const int fb   = ft * 16;

    const float w = 1.0f - alpha_p[0];
    float dec[8];
    {
        float p = powi_f(w, 8 * half + 1);
        #pragma unroll
        for (int j = 0; j < 8; ++j) { dec[j] = p; p *= w; }
    }
    const float w16 = powi_f(w, 16);

    float cur = seg_in[(((size_t)b * NFT + ft) * NSEG + seg) * 16 + m];

    for (int c = 0; c < CHUNKS; ++c) {
        if (__ballot(cur != 0.0f) == 0ull) break;
        const int t0 = seg * SEGLEN + c * 16;
        const size_t base = ((size_t)b * SEQ + t0) * FEAT + fb + m;
        #pragma unroll
        for (int j = 0; j < 8; ++j) {
            const size_t idx = base + (size_t)(j + 8 * half) * FEAT;
            out[idx] += dec[j] * cur;
        }
        cur *= w16;
    }
}

extern "C" void kernel_launch(void* const* d_in, const int* in_sizes, int n_in,
                              void* d_out, int out_size, void* d_ws, size_t ws_size,
                              hipStream_t stream) {
    const float* x     = (const float*)d_in[0];
    const float* alpha = (const float*)d_in[1];
    float* out = (float*)d_out;

    float* carries = (float*)d_ws;                                   // 512 KB
    float* seg_in  = carries + (size_t)BATCH * NFT * NSEG * 16;      // 512 KB

    const dim3 blk(128);                 // 4 waves/block
    const dim3 grd(WAVES / 4);           // 2048 blocks

    ewma_scan_kernel<<<grd, blk, 0, stream>>>(x, alpha, out, carries);
    ewma_combine_kernel<<<(BATCH * FEAT + 255) / 256, 256, 0, stream>>>(carries, alpha, seg_in);
    ewma_fixup_kernel<<<grd, blk, 0, stream>>>(seg_in, alpha, out);
}